// RecursiveMamba130M_43250320670982
// MI455X (gfx1250) — compile-verified
//
#include <hip/hip_runtime.h>
#include <math.h>

// ---------------------------------------------------------------------------
// RecursiveMamba block for MI455X (gfx1250, wave32).
// Shapes (fixed by the reference): B=2, L=1024, D=768, NC=2, S=16, E=NC*D=1536.
//
// Roofline: state stream = 805MB HBM (~35us @ 23.3TB/s). GEMMs = 9.7 GFLOP
// fp32 on V_WMMA_F32_16X16X4_F32 with all operands L2-resident; 64x64
// register blocking gives 16 FLOP/byte from L2 (~300MB L2 traffic per GEMM).
// GEMM waves run 1-per-SIMD with a raised VGPR budget (launch_bounds(128,1))
// so the 4x4 accumulator block lives entirely in registers (no scratch).
// State tensors use non-temporal (TH=NT) hints so they don't evict the
// L2-resident GEMM operands.
// ---------------------------------------------------------------------------

typedef __attribute__((ext_vector_type(2))) float v2f;
typedef __attribute__((ext_vector_type(4))) float v4f;
typedef __attribute__((ext_vector_type(8))) float v8f;

#define EPS_F 1.1920928955078125e-07f

constexpr int kB  = 2;
constexpr int kL  = 1024;
constexpr int kD  = 768;
constexpr int kNC = 2;
constexpr int kS  = 16;
constexpr int kE  = kNC * kD;        // 1536
constexpr int kBL = kB * kL;         // 2048
constexpr int kTab = kNC * kD * kS;  // 24576

// ---------------------------------------------------------------------------
// K0: cos/sin tables for A_theta (shared across all (b,l) positions).
// ---------------------------------------------------------------------------
__global__ void k_sincos(const float* __restrict__ theta,
                         float* __restrict__ cosT, float* __restrict__ sinT) {
  int i = blockIdx.x * blockDim.x + threadIdx.x;
  if (i < kTab) {
    float sv, cv;
    sincosf(theta[i], &sv, &cv);
    cosT[i] = cv;
    sinT[i] = sv;
  }
}

// ---------------------------------------------------------------------------
// K1/K3: C[M,N] = A[M,K] * Bt[N,K]^T   (both row-major over K; fp32 WMMA).
// One wave owns a 64x64 macro-tile = 4x4 grid of 16x16 WMMA tiles.
// Per K-step: 4 A-frag + 4 B-frag b64 loads feed 16 WMMAs (16 FLOP/byte).
// Fragment row offsets (i*16*K) are compile-time constants folded into the
// 24-bit instruction offset, so only two base pointers stay live.
// 128-thread blocks (1 wave/SIMD) + launch_bounds(128,1): the 128-VGPR
// accumulator block stays in registers; the XDL pipe is fed by 16
// independent WMMA chains (ISA 7.5).
//
// Fragment layouts (CDNA5 ISA 7.12.2, f32 16x16x4):
//   A 16x4 : lane holds row M = lane&15; VGPR v holds K = 2*(lane>>4) + v
//   B 4x16 : lane holds col N = lane&15; VGPR v holds K = 2*(lane>>4) + v
//   C 16x16: lane holds col N = lane&15; VGPR j holds row M = j + 8*(lane>>4)
// Grid is sized so every wave is fully active (EXEC all ones, as WMMA needs).
// ---------------------------------------------------------------------------
template <int M, int N, int K>
__global__ void __launch_bounds__(128, 1)
k_gemm_nt(const float* __restrict__ A,
          const float* __restrict__ Bt,
          float* __restrict__ C) {
  constexpr int TN = N / 64;
  const int wave = blockIdx.x * (blockDim.x >> 5) + (threadIdx.x >> 5);
  const int tm = (wave / TN) * 64;
  const int tn = (wave % TN) * 64;
  const int lane = threadIdx.x & 31;
  const int r  = lane & 15;          // M row (A frag) / N col (B frag)
  const int kh = (lane >> 4) << 1;   // K sub-offset {0,2}

  const float* __restrict__ aBase = A  + (size_t)(tm + r) * K + kh;
  const float* __restrict__ bBase = Bt + (size_t)(tn + r) * K + kh;

  v8f acc[4][4] = {};
#pragma unroll 2
  for (int k = 0; k < K; k += 4) {
    v2f a[4], b[4];
#pragma unroll
    for (int i = 0; i < 4; ++i) {
      a[i] = *(const v2f*)(aBase + k + i * 16 * K);  // const ioffset
      b[i] = *(const v2f*)(bBase + k + i * 16 * K);  // const ioffset
    }
#pragma unroll
    for (int im = 0; im < 4; ++im)
#pragma unroll
      for (int in = 0; in < 4; ++in)
        acc[im][in] = __builtin_amdgcn_wmma_f32_16x16x4_f32(
            /*neg_a=*/false, a[im], /*neg_b=*/false, b[in],
            /*c_mod=*/(short)0, acc[im][in],
            /*reuse_a=*/false, /*reuse_b=*/false);
  }

  // Store: one base address, 128 compile-time offsets.
  const int mb = tm + ((lane >> 4) << 3);  // row base within macro-tile
  float* __restrict__ cBase = C + (size_t)mb * N + tn + r;
#pragma unroll
  for (int im = 0; im < 4; ++im)
#pragma unroll
    for (int in = 0; in < 4; ++in)
#pragma unroll
      for (int j = 0; j < 8; ++j)
        cBase[(size_t)(16 * im + j) * N + 16 * in] = acc[im][in][j];
}

// ---------------------------------------------------------------------------
// K2: fused state rotation + injection + C-readout.
// 4 threads per (b,l,nc,d) row; each thread handles 4 of the S=16 states with
// 128-bit non-temporal loads/stores (state is read-once/write-once; keep it
// out of the L2 so GEMM operands stay resident). y reduced across the 4
// lanes via shfl_xor.
// ---------------------------------------------------------------------------
__global__ void k_state(const float* __restrict__ x_in,
                        const float* __restrict__ rs,
                        const float* __restrict__ is,
                        const float* __restrict__ cosT,
                        const float* __restrict__ sinT,
                        const float* __restrict__ Br,
                        const float* __restrict__ Bi,
                        const float* __restrict__ Cr,
                        const float* __restrict__ Ci,
                        float* __restrict__ nr_out,
                        float* __restrict__ ni_out,
                        float* __restrict__ y_flat) {
  const int T   = blockIdx.x * blockDim.x + threadIdx.x;
  const int q   = T & 3;          // quarter of S
  const int row = T >> 2;         // = bl*kE + e
  const int e   = row % kE;
  const int bl  = row / kE;

  const size_t sbase = (size_t)row * kS + q * 4;
  const int    tIdx  = e * kS + q * 4;

  const v4f r4 = __builtin_nontemporal_load((const v4f*)(rs + sbase));
  const v4f i4 = __builtin_nontemporal_load((const v4f*)(is + sbase));
  const v4f c4  = *(const v4f*)(cosT + tIdx);
  const v4f s4  = *(const v4f*)(sinT + tIdx);
  const v4f b4r = *(const v4f*)(Br + tIdx);
  const v4f b4i = *(const v4f*)(Bi + tIdx);
  const v4f c4r = *(const v4f*)(Cr + tIdx);
  const v4f c4i = *(const v4f*)(Ci + tIdx);
  const float xv = x_in[(size_t)bl * kE + e];

  v4f nr, ni;
  nr.x = fmaf(c4.x, r4.x, fmaf(-s4.x, i4.x, b4r.x * xv));
  nr.y = fmaf(c4.y, r4.y, fmaf(-s4.y, i4.y, b4r.y * xv));
  nr.z = fmaf(c4.z, r4.z, fmaf(-s4.z, i4.z, b4r.z * xv));
  nr.w = fmaf(c4.w, r4.w, fmaf(-s4.w, i4.w, b4r.w * xv));
  ni.x = fmaf(s4.x, r4.x, fmaf(c4.x, i4.x, b4i.x * xv));
  ni.y = fmaf(s4.y, r4.y, fmaf(c4.y, i4.y, b4i.y * xv));
  ni.z = fmaf(s4.z, r4.z, fmaf(c4.z, i4.z, b4i.z * xv));
  ni.w = fmaf(s4.w, r4.w, fmaf(c4.w, i4.w, b4i.w * xv));

  __builtin_nontemporal_store(nr, (v4f*)(nr_out + sbase));
  __builtin_nontemporal_store(ni, (v4f*)(ni_out + sbase));

  float y = fmaf(c4r.x, nr.x, -c4i.x * ni.x)
          + fmaf(c4r.y, nr.y, -c4i.y * ni.y)
          + fmaf(c4r.z, nr.z, -c4i.z * ni.z)
          + fmaf(c4r.w, nr.w, -c4i.w * ni.w);
  y += __shfl_xor(y, 1, 32);
  y += __shfl_xor(y, 2, 32);
  if (q == 0) y_flat[(size_t)bl * kE + e] = y;
}

// ---------------------------------------------------------------------------
// K4: RMSNorm over D + residual add. One 256-thread block (8 waves) per row.
// ---------------------------------------------------------------------------
__global__ void k_rmsnorm_res(const float* __restrict__ x,
                              const float* __restrict__ pre,
                              const float* __restrict__ w,
                              float* __restrict__ out) {
  const int bl = blockIdx.x;
  const float* __restrict__ row = pre + (size_t)bl * kD;

  float ss = 0.f;
  for (int i = threadIdx.x; i < kD; i += blockDim.x) {
    float v = row[i];
    ss = fmaf(v, v, ss);
  }
#pragma unroll
  for (int o = 16; o >= 1; o >>= 1) ss += __shfl_xor(ss, o, 32);

  __shared__ float red[8];
  if ((threadIdx.x & 31) == 0) red[threadIdx.x >> 5] = ss;
  __syncthreads();
  float tot = 0.f;
#pragma unroll
  for (int wv = 0; wv < 8; ++wv) tot += red[wv];

  const float scale = rsqrtf(tot * (1.0f / (float)kD) + EPS_F);
  for (int i = threadIdx.x; i < kD; i += blockDim.x) {
    out[(size_t)bl * kD + i] =
        x[(size_t)bl * kD + i] + row[i] * scale * w[i];
  }
}

// ---------------------------------------------------------------------------
extern "C" void kernel_launch(void* const* d_in, const int* in_sizes, int n_in,
                              void* d_out, int out_size, void* d_ws, size_t ws_size,
                              hipStream_t stream) {
  const float* x     = (const float*)d_in[0];
  const float* rs    = (const float*)d_in[1];
  const float* is    = (const float*)d_in[2];
  const float* W_in  = (const float*)d_in[3];
  const float* A_th  = (const float*)d_in[4];
  const float* Br    = (const float*)d_in[5];
  const float* Bi    = (const float*)d_in[6];
  const float* Cr    = (const float*)d_in[7];
  const float* Ci    = (const float*)d_in[8];
  const float* W_out = (const float*)d_in[9];
  const float* nw    = (const float*)d_in[10];

  // Outputs concatenated in tuple order: (x+out), new_real, new_imag.
  float* out = (float*)d_out;
  float* nr  = out + (size_t)kBL * kD;
  float* ni  = nr + (size_t)kBL * kNC * kD * kS;

  // Workspace layout (floats): cosT | sinT | x_in | y_flat ; out_pre reuses x_in.
  float* ws   = (float*)d_ws;
  float* cosT = ws;
  float* sinT = cosT + kTab;
  float* x_in = sinT + kTab;
  float* yfl  = x_in + (size_t)kBL * kE;
  float* pre  = x_in;  // x_in is dead after k_state

  // K0: tables
  k_sincos<<<(kTab + 255) / 256, 256, 0, stream>>>(A_th, cosT, sinT);

  // K1: x_in = X * W_in^T   (2048x1536); 64x64 macro-tiles: 768 waves, 4/block
  k_gemm_nt<kBL, kE, kD>
      <<<(kBL / 64) * (kE / 64) / 4, 128, 0, stream>>>(x, W_in, x_in);

  // K2: state rotation + readout (the 805MB HBM stream, NT-hinted)
  const int stThreads = kBL * kE * 4;           // 12,582,912 (exact /256)
  k_state<<<stThreads / 256, 256, 0, stream>>>(x_in, rs, is, cosT, sinT,
                                               Br, Bi, Cr, Ci, nr, ni, yfl);

  // K3: out_pre = y_flat * W_out^T   (2048x768); 384 waves, 4/block
  k_gemm_nt<kBL, kD, kE>
      <<<(kBL / 64) * (kD / 64) / 4, 128, 0, stream>>>(yfl, W_out, pre);

  // K4: RMSNorm + residual
  k_rmsnorm_res<<<kBL, 256, 0, stream>>>(x, pre, nw, out);
}